// trunc_conv_73985106641584
// MI455X (gfx1250) — compile-verified
//
#include <hip/hip_runtime.h>

typedef __attribute__((ext_vector_type(2))) float v2f;
typedef __attribute__((ext_vector_type(8))) float v8f;
typedef unsigned int u32x4 __attribute__((ext_vector_type(4)));
typedef int i32x4 __attribute__((ext_vector_type(4)));
typedef int i32x8 __attribute__((ext_vector_type(8)));

// ---------------- problem constants ----------------
constexpr int Bn    = 32;           // batch
constexpr int CIN   = 32;           // input channels
constexpr int L     = 64;           // spatial
constexpr int COUT  = 64;           // output channels
constexpr int TOPK  = 512;
constexpr int OD    = 62;           // L - 3 + 1
constexpr int KDIM  = CIN * 9;      // 288 GEMM K
constexpr int LP    = L + 2;        // 66 (pad=1 each side)
constexpr int XPLANE = L * L;       // 4096
constexpr int XIMG   = CIN * XPLANE;   // 131072 elements per batch
constexpr int NTOT   = Bn * XIMG;      // 4194304
constexpr int XZP_PER_B = CIN * LP * LP;  // 139392
constexpr int XZP_TOT   = Bn * XZP_PER_B; // 4460544

// ---------------- workspace layout (in 4-byte elements) ----------------
constexpr int OFF_WSUM   = 0;                       // 288 f
constexpr int OFF_KASUM  = 512;                     // 32*62*62 = 123008 f
constexpr int OFF_TERM2  = OFF_KASUM + 123008;      // 131072 f
constexpr int OFF_KEYS   = OFF_TERM2 + 131072;      // 4194304 u32
constexpr int OFF_THRESH = OFF_KEYS + NTOT;         // 64 u32
constexpr int OFF_QUOTA  = OFF_THRESH + 64;         // 64 i32
constexpr int OFF_XZP    = OFF_QUOTA + 256;         // 4460544 f

// ============================================================
// 1) wsum[c,u,v] = sum_o w[o,c,u,v]      (288 values)
// ============================================================
__global__ void wsum_kernel(const float* __restrict__ w, float* __restrict__ wsum) {
    int k = threadIdx.x;
    if (k >= KDIM) return;
    float s = 0.f;
    for (int o = 0; o < COUT; ++o) s += w[o * KDIM + k];
    wsum[k] = s;
}

// ============================================================
// 2) ka_sum[b,p,q] = (1/288) sum_{c,u,v} x[b,c,p+u,q+v]*wsum[c,u,v]
// ============================================================
__global__ void kasum_kernel(const float* __restrict__ x, const float* __restrict__ wsum,
                             float* __restrict__ kasum) {
    int idx = blockIdx.x * blockDim.x + threadIdx.x;
    if (idx >= Bn * OD * OD) return;
    int b = idx / (OD * OD);
    int pq = idx % (OD * OD);
    int p = pq / OD, q = pq % OD;
    const float* xb = x + b * XIMG + p * L + q;
    float s = 0.f;
    for (int c = 0; c < CIN; ++c)
        for (int u = 0; u < 3; ++u)
            for (int v = 0; v < 3; ++v)
                s += xb[c * XPLANE + u * L + v] * wsum[c * 9 + u * 3 + v];
    kasum[idx] = s * (1.0f / 288.0f);
}

// ============================================================
// 3) term2[b,i,j] = sum of ka_sum over {i-2..i}x{j-2..j} clipped to [0,62)
// ============================================================
__global__ void term2_kernel(const float* __restrict__ kasum, float* __restrict__ term2) {
    int idx = blockIdx.x * blockDim.x + threadIdx.x;
    if (idx >= Bn * XPLANE) return;
    int b = idx >> 12;
    int ij = idx & 4095;
    int i = ij >> 6, j = ij & 63;
    float s = 0.f;
    for (int p = i - 2; p <= i; ++p) {
        if (p < 0 || p >= OD) continue;
        for (int q = j - 2; q <= j; ++q) {
            if (q < 0 || q >= OD) continue;
            s += kasum[b * (OD * OD) + p * OD + q];
        }
    }
    term2[idx] = s;
}

// ============================================================
// 4) score r -> order-preserving u32 key
// ============================================================
__global__ void score_kernel(const float* __restrict__ x, const float* __restrict__ wsum,
                             const float* __restrict__ term2, unsigned* __restrict__ keys) {
    int idx = blockIdx.x * blockDim.x + threadIdx.x;
    if (idx >= NTOT) return;
    int b = idx >> 17;
    int rem = idx & (XIMG - 1);
    int c = rem >> 12;
    int ij = rem & 4095;
    int i = ij >> 6, j = ij & 63;
    int ulo = (i > OD - 1) ? (i - (OD - 1)) : 0;
    int uhi = (i < 2) ? i : 2;
    int vlo = (j > OD - 1) ? (j - (OD - 1)) : 0;
    int vhi = (j < 2) ? j : 2;
    float colsum = 0.f;
    for (int u = ulo; u <= uhi; ++u)
        for (int v = vlo; v <= vhi; ++v)
            colsum += wsum[c * 9 + u * 3 + v];
    int ci = (i == 0 || i == 63) ? 1 : ((i == 1 || i == 62) ? 2 : 3);
    int cj = (j == 0 || j == 63) ? 1 : ((j == 1 || j == 62) ? 2 : 3);
    float cnt = (float)(ci * cj * COUT);
    float r = (x[idx] * colsum - term2[b * XPLANE + ij]) * cnt;
    unsigned u32 = __float_as_uint(r);
    keys[idx] = (u32 & 0x80000000u) ? ~u32 : (u32 | 0x80000000u);
}

// ============================================================
// 5) 4-pass MSB radix select: Kth largest key per (batch, dir)
// ============================================================
__global__ __launch_bounds__(256) void radix_select_kernel(const unsigned* __restrict__ keys,
                                                           unsigned* __restrict__ thresh,
                                                           int* __restrict__ quota) {
    int batch = blockIdx.x >> 1;
    int dir   = blockIdx.x & 1;
    unsigned flip = dir ? 0xFFFFFFFFu : 0u;
    const unsigned* kb = keys + batch * XIMG;
    __shared__ unsigned hist[256];
    __shared__ unsigned s_prefix;
    __shared__ int s_K;
    if (threadIdx.x == 0) { s_prefix = 0u; s_K = TOPK; }
    __syncthreads();
    for (int pass = 3; pass >= 0; --pass) {
        hist[threadIdx.x] = 0u;
        __syncthreads();
        unsigned prefix = s_prefix;
        unsigned mask = (pass == 3) ? 0u : (0xFFFFFFFFu << ((pass + 1) * 8));
        int sh = pass * 8;
        for (int idx = threadIdx.x; idx < XIMG; idx += 256) {
            unsigned key = kb[idx] ^ flip;
            if ((key & mask) == (prefix & mask))
                atomicAdd(&hist[(key >> sh) & 255u], 1u);
        }
        __syncthreads();
        if (threadIdx.x == 0) {
            int K = s_K, cum = 0, bin = 255;
            for (; bin > 0; --bin) {
                int h = (int)hist[bin];
                if (cum + h >= K) break;
                cum += h;
            }
            s_prefix = prefix | ((unsigned)bin << sh);
            s_K = K - cum;
        }
        __syncthreads();
    }
    if (threadIdx.x == 0) {
        thresh[blockIdx.x] = s_prefix;  // flipped domain for dir=1
        quota[blockIdx.x]  = s_K;       // # ties at threshold still selected
    }
}

// ============================================================
// 6a) zero the padded xz buffer
// ============================================================
__global__ void zero_kernel(float* __restrict__ p, int n) {
    int idx = blockIdx.x * blockDim.x + threadIdx.x;
    if (idx < n) p[idx] = 0.f;
}

// ============================================================
// 6b) apply mask, write x*z into zero-padded layout [b][c][66][66]
// ============================================================
__global__ void mask_apply_kernel(const float* __restrict__ x, const unsigned* __restrict__ keys,
                                  const unsigned* __restrict__ thresh, int* __restrict__ quota,
                                  float* __restrict__ xzp) {
    int idx = blockIdx.x * blockDim.x + threadIdx.x;
    if (idx >= NTOT) return;
    int b = idx >> 17;
    int rem = idx & (XIMG - 1);
    unsigned key = keys[idx];
    float z = 1.f;
    unsigned Tt = thresh[2 * b];
    if (key > Tt) z = 0.f;
    else if (key == Tt) { if (atomicSub(&quota[2 * b], 1) > 0) z = 0.f; }
    unsigned kf = ~key;
    unsigned Tb = thresh[2 * b + 1];
    if (kf > Tb) z = 0.f;
    else if (kf == Tb) { if (atomicSub(&quota[2 * b + 1], 1) > 0) z = 0.f; }
    int c = rem >> 12;
    int ij = rem & 4095;
    int i = ij >> 6, j = ij & 63;
    xzp[b * XZP_PER_B + c * (LP * LP) + (i + 1) * LP + (j + 1)] = x[idx] * z;
}

// ============================================================
// 7) main conv as GEMM via V_WMMA_F32_16X16X4_F32
//    Each wave: D(16 outch x 16 j) for BOTH otiles of its half,
//    sharing one B fragment (x-patches) per K-step.
//    Weights staged into LDS (stride 289) by the Tensor Data Mover.
//    grid: (1024, 2) x 256; blockIdx.y selects the 32-outch half.
// ============================================================
__global__ __launch_bounds__(256) void conv_wmma_kernel(const float* __restrict__ xzp,
                                                        const float* __restrict__ w,
                                                        const float* __restrict__ bias,
                                                        float* __restrict__ out) {
    constexpr int WSTRIDE = KDIM + 1;  // 289: odd stride -> conflict-free ds reads
    __shared__ float Wlds[32 * WSTRIDE];
    __shared__ int   offA[KDIM];

    int h = blockIdx.y;  // out-channel half (0/1)

    // ---- TDM: iterate-mode load of 32 rows x 288 dwords from w+h*32*288,
    //      LDS stride 289 dwords (tile = 1 row; 32 iterations).
    if (threadIdx.x == 0) {
        unsigned lds_base = (unsigned)(unsigned long long)(&Wlds[0]);  // LDS aperture low 32 = offset
        unsigned long long ga = (unsigned long long)(const void*)(w + (size_t)h * 32 * KDIM);
        u32x4 g0;
        g0[0] = 1u;                                            // count=1 valid descriptor
        g0[1] = lds_base;                                      // lds_addr (bytes)
        g0[2] = (unsigned)ga;                                  // global_addr[31:0]
        g0[3] = (unsigned)((ga >> 32) & 0x1FFFFFFu) | (2u << 30);  // addr[56:32] | type=2
        i32x8 g1;
        g1[0] = (2 << 16) | (1 << 19);   // data_size=4B, iterate_enable=1
        g1[1] = (KDIM & 0xFFFF) << 16;   // tensor_dim0 = 288 (dwords)
        g1[2] = 32 << 16;                // tensor_dim1 = 32
        g1[3] = KDIM << 16;              // tile_dim0 = 288
        g1[4] = 1;                       // tile_dim1 = 1, tile_dim2 = 0
        g1[5] = KDIM;                    // tensor_dim0_stride = 288
        g1[6] = 0;
        g1[7] = 0;
        i32x4 g2;
        g2[0] = 0;                       // tensor_dim2 unused
        g2[1] = WSTRIDE;                 // lds_addr_increment = 289 (data_size units)
        g2[2] = KDIM;                    // global_addr_increment = 288 (data_size units)
        g2[3] = 31 << 16;                // iterate_count = 31 -> 32 iterations
        i32x4 g3 = {0, 0, 0, 0};
#if __clang_major__ >= 23
        i32x8 g4 = {0, 0, 0, 0, 0, 0, 0, 0};
        __builtin_amdgcn_tensor_load_to_lds(g0, g1, g2, g3, g4, 0);
#else
        __builtin_amdgcn_tensor_load_to_lds(g0, g1, g2, g3, 0);
#endif
    }
    // im2col dword offsets into padded image: k = c*9+u*3+v
    for (int k = threadIdx.x; k < KDIM; k += 256) {
        int c = k / 9, rm = k % 9, u = rm / 3, v = rm % 3;
        offA[k] = c * (LP * LP) + u * LP + v;
    }
    __builtin_amdgcn_s_wait_tensorcnt(0);
    __syncthreads();

    int wave = threadIdx.x >> 5;
    int lane = threadIdx.x & 31;
    int s = blockIdx.x * 8 + wave;               // spatial tile id in [0,8192)
    int j0 = (s & 3) * 16;
    int i  = (s >> 2) & 63;
    int b  = s >> 8;

    int nl    = lane & 15;
    int half2 = (lane >> 4) << 1;                // +2 K-offset for upper half-wave

    const float* xb    = xzp + b * XZP_PER_B + i * LP + (j0 + nl);  // N = j0+nl column
    const float* a0row = &Wlds[nl * WSTRIDE];                       // otile 0 row
    const float* a1row = &Wlds[(16 + nl) * WSTRIDE];                // otile 1 row

    v8f acc0 = {0.f, 0.f, 0.f, 0.f, 0.f, 0.f, 0.f, 0.f};
    v8f acc1 = {0.f, 0.f, 0.f, 0.f, 0.f, 0.f, 0.f, 0.f};
    for (int k0 = 0; k0 < KDIM; k0 += 4) {
        int ka = k0 + half2;
        v2f a0, a1, bm;
        bm.x = xb[offA[ka]];
        bm.y = xb[offA[ka + 1]];
        a0.x = a0row[ka];
        a0.y = a0row[ka + 1];
        a1.x = a1row[ka];
        a1.y = a1row[ka + 1];
        acc0 = __builtin_amdgcn_wmma_f32_16x16x4_f32(
            false, a0, false, bm, (short)0, acc0, false, false);
        acc1 = __builtin_amdgcn_wmma_f32_16x16x4_f32(
            false, a1, false, bm, (short)0, acc1, false, false);
    }

    // D layout: lane -> N = lane%16 (= j), VGPR r -> M = r (+8 upper half-wave)
    int jj = j0 + nl;
    int ob0 = h * 32 + ((lane >> 4) << 3);        // otile 0
    int ob1 = ob0 + 16;                           // otile 1
#pragma unroll
    for (int r = 0; r < 8; ++r) {
        int o0 = ob0 + r;
        out[((b * COUT + o0) * L + i) * L + jj] = acc0[r] + bias[o0];
    }
#pragma unroll
    for (int r = 0; r < 8; ++r) {
        int o1 = ob1 + r;
        out[((b * COUT + o1) * L + i) * L + jj] = acc1[r] + bias[o1];
    }
}

// ============================================================
extern "C" void kernel_launch(void* const* d_in, const int* in_sizes, int n_in,
                              void* d_out, int out_size, void* d_ws, size_t ws_size,
                              hipStream_t stream) {
    const float* x    = (const float*)d_in[0];
    const float* w    = (const float*)d_in[1];
    const float* bias = (const float*)d_in[2];
    float* out = (float*)d_out;
    float* ws  = (float*)d_ws;

    float*    wsum   = ws + OFF_WSUM;
    float*    kasum  = ws + OFF_KASUM;
    float*    term2  = ws + OFF_TERM2;
    unsigned* keys   = (unsigned*)(ws + OFF_KEYS);
    unsigned* thresh = (unsigned*)(ws + OFF_THRESH);
    int*      quota  = (int*)(ws + OFF_QUOTA);
    float*    xzp    = ws + OFF_XZP;

    wsum_kernel<<<1, 288, 0, stream>>>(w, wsum);
    kasum_kernel<<<(Bn * OD * OD + 255) / 256, 256, 0, stream>>>(x, wsum, kasum);
    term2_kernel<<<(Bn * XPLANE + 255) / 256, 256, 0, stream>>>(kasum, term2);
    score_kernel<<<NTOT / 256, 256, 0, stream>>>(x, wsum, term2, keys);
    radix_select_kernel<<<Bn * 2, 256, 0, stream>>>(keys, thresh, quota);
    zero_kernel<<<(XZP_TOT + 255) / 256, 256, 0, stream>>>(xzp, XZP_TOT);
    mask_apply_kernel<<<NTOT / 256, 256, 0, stream>>>(x, keys, thresh, quota, xzp);
    conv_wmma_kernel<<<dim3(1024, 2), 256, 0, stream>>>(xzp, w, bias, out);
}